// TiDHy_2405181686597
// MI455X (gfx1250) — compile-verified
//
#include <hip/hip_runtime.h>
#include <math.h>

// ---------------- problem constants (match reference) ----------------
#define R_DIM    128
#define R2_DIM   64
#define MIXK     32
#define IN_DIM   512
#define HID      256
#define BB       256
#define TT       16
#define MAX_ITERS 12
#define SPW      16               // samples per workgroup (== WMMA M)
#define NWG      (BB / SPW)       // 16 workgroups
#define NTHR     512              // 16 waves of 32
#define NW       (NTHR / 32)
#define LN_EPS   1e-6f

// ws layout (floats): [0..48) per-WG loss sums; P at WS_P_OFF; transposed weights after
#define WS_P_OFF  64
#define WS_P_SIZE (NWG * SPW * MIXK * R_DIM)        // 1,048,576 floats
#define WS_WT_OFF (WS_P_OFF + WS_P_SIZE)
#define DECWT_SZ  (IN_DIM * R_DIM)                  // 65536
#define H1WT_SZ   (HID * R2_DIM)                    // 16384
#define H2WT_SZ   (HID * HID)                       // 65536
#define H3WT_SZ   (MIXK * HID)                      // 8192

typedef __attribute__((ext_vector_type(2))) float v2f;
typedef __attribute__((ext_vector_type(8))) float v8f;

// ---------------- WMMA f32 16x16 tile helper (K stepped by 4) ----------------
// A: 16 x K row-major in LDS (lda). Per ISA layout: lane L holds A[L&15][k0+2*(L>>4) .. +1]
// B: logical (K x 16) tile taken from WT stored (N x K) row-major:
//    B[k][c] = WT[(col0+c)*ldw + k]  -> k-contiguous, one global_load_b64 per fragment
__device__ __forceinline__ v8f wmma_tile(const float* __restrict__ lA, int lda,
                                         const float* __restrict__ WT, int ldw,
                                         int K, int col0, int lane) {
  const int rn = lane & 15;
  const int kh = (lane >> 4) << 1;      // 0 or 2
  const float* Wc = WT + (size_t)(col0 + rn) * ldw + kh;
  v8f acc = {};
  for (int k0 = 0; k0 < K; k0 += 4) {
    v2f a, b;
    a.x = lA[rn * lda + k0 + kh];
    a.y = lA[rn * lda + k0 + kh + 1];
    b.x = Wc[k0];
    b.y = Wc[k0 + 1];
    acc = __builtin_amdgcn_wmma_f32_16x16x4_f32(false, a, false, b,
                                                (short)0, acc, false, false);
  }
  return acc;
}

__device__ __forceinline__ float wave_sum(float v) {
#pragma unroll
  for (int off = 16; off > 0; off >>= 1) v += __shfl_down(v, off, 32);
  return v;
}

__device__ __forceinline__ float soft_thresh(float v, float lmda) {
  float a = fabsf(v) - lmda;
  a = a > 0.f ? a : 0.f;
  return v >= 0.f ? a : -a;    // relu(|v|-l)*sign(v)
}

// ---------------- shared-memory layout + weight bundle ----------------
struct Sm {
  float *xt, *ebuf, *r, *rprev, *rhat, *gsp, *r2, *gr2;
  float *x1, *a1, *x2d, *wmix, *x3, *dwm, *stats;
};
struct Wt {
  const float *decW, *decb, *temporal, *h1W, *h1b, *lns, *lnb, *h2W, *h2b, *h3W, *h3b;
  const float *decWT, *h1WT, *h2WT, *h3WT;   // transposed copies in ws (all k-contiguous)
};

// ---------------- hypernet forward: r2 -> wmix (saves x1, a1, x2, x3, LN stats) --
__device__ void hypernet_fwd(const Sm& S, const Wt& W, int tid, int lane, int wave) {
  // x1 = r2 @ h1W + h1b   (16 x 256, K=64)  B from h1WT (HID x R2)
  for (int tile = wave; tile < HID / 16; tile += NW) {
    int c0 = tile * 16;
    v8f acc = wmma_tile(S.r2, R2_DIM, W.h1WT, R2_DIM, R2_DIM, c0, lane);
    int rn = lane & 15, sb = (lane >> 4) * 8, c = c0 + rn;
#pragma unroll
    for (int v = 0; v < 8; ++v) S.x1[(sb + v) * HID + c] = acc[v] + W.h1b[c];
  }
  __syncthreads();
  // LayerNorm stats per sample
  if (tid < SPW) {
    const float* xr = S.x1 + tid * HID;
    float mu = 0.f;
    for (int h = 0; h < HID; ++h) mu += xr[h];
    mu *= (1.f / HID);
    float var = 0.f;
    for (int h = 0; h < HID; ++h) { float d = xr[h] - mu; var += d * d; }
    var *= (1.f / HID);
    S.stats[tid] = mu;
    S.stats[16 + tid] = rsqrtf(var + LN_EPS);
  }
  __syncthreads();
  // a1 = elu(LN(x1)*ln_s + ln_b)
  for (int i = tid; i < SPW * HID; i += NTHR) {
    int s = i / HID, h = i % HID;
    float xh = (S.x1[i] - S.stats[s]) * S.stats[16 + s];
    float v = xh * W.lns[h] + W.lnb[h];
    S.a1[i] = v > 0.f ? v : (__expf(v) - 1.f);
  }
  __syncthreads();
  // x2 = a1 @ h2W + h2b   (16 x 256, K=256)  B from h2WT (HID x HID)
  for (int tile = wave; tile < HID / 16; tile += NW) {
    int c0 = tile * 16;
    v8f acc = wmma_tile(S.a1, HID, W.h2WT, HID, HID, c0, lane);
    int rn = lane & 15, sb = (lane >> 4) * 8, c = c0 + rn;
#pragma unroll
    for (int v = 0; v < 8; ++v) S.x2d[(sb + v) * HID + c] = acc[v] + W.h2b[c];
  }
  __syncthreads();
  // x3 = x2 @ h3W + h3b ; wmix = relu(x3)   (16 x 32, K=256)  B from h3WT (MIXK x HID)
  for (int tile = wave; tile < MIXK / 16; tile += NW) {
    int c0 = tile * 16;
    v8f acc = wmma_tile(S.x2d, HID, W.h3WT, HID, HID, c0, lane);
    int rn = lane & 15, sb = (lane >> 4) * 8, c = c0 + rn;
#pragma unroll
    for (int v = 0; v < 8; ++v) {
      float xv = acc[v] + W.h3b[c];
      S.x3[(sb + v) * MIXK + c] = xv;
      S.wmix[(sb + v) * MIXK + c] = xv > 0.f ? xv : 0.f;
    }
  }
  __syncthreads();
}

// ---------------- hypernet backward: dwm (dE/dw) -> gr2 ----------------
__device__ void hypernet_bwd(const Sm& S, const Wt& W, int tid, int lane, int wave) {
  // dx3 = dwm * 1[x3>0] (relu)
  for (int i = tid; i < SPW * MIXK; i += NTHR)
    S.dwm[i] = S.x3[i] > 0.f ? S.dwm[i] : 0.f;
  __syncthreads();
  // dx2 = dx3 @ h3W^T  (K=32) -> x2d ; B from original h3W (HID x MIXK), k-contiguous
  for (int tile = wave; tile < HID / 16; tile += NW) {
    int c0 = tile * 16;
    v8f acc = wmma_tile(S.dwm, MIXK, W.h3W, MIXK, MIXK, c0, lane);
    int rn = lane & 15, sb = (lane >> 4) * 8, c = c0 + rn;
#pragma unroll
    for (int v = 0; v < 8; ++v) S.x2d[(sb + v) * HID + c] = acc[v];
  }
  __syncthreads();
  // da1 = dx2 @ h2W^T  (K=256) -> ebuf ; B from original h2W
  for (int tile = wave; tile < HID / 16; tile += NW) {
    int c0 = tile * 16;
    v8f acc = wmma_tile(S.x2d, HID, W.h2W, HID, HID, c0, lane);
    int rn = lane & 15, sb = (lane >> 4) * 8, c = c0 + rn;
#pragma unroll
    for (int v = 0; v < 8; ++v) S.ebuf[(sb + v) * HID + c] = acc[v];
  }
  __syncthreads();
  // dxhat = da1 * elu'(ln_out) * ln_s   (elu' = a1>0 ? 1 : a1+1)
  for (int i = tid; i < SPW * HID; i += NTHR) {
    int h = i % HID;
    float ep = S.a1[i] > 0.f ? 1.f : (S.a1[i] + 1.f);
    S.ebuf[i] = S.ebuf[i] * ep * W.lns[h];
  }
  __syncthreads();
  // LN backward stats
  if (tid < SPW) {
    const float* dx = S.ebuf + tid * HID;
    const float* xp = S.x1 + tid * HID;
    float mu = S.stats[tid], rstd = S.stats[16 + tid];
    float m1 = 0.f, m2 = 0.f;
    for (int h = 0; h < HID; ++h) {
      float xh = (xp[h] - mu) * rstd;
      m1 += dx[h];
      m2 += dx[h] * xh;
    }
    S.stats[32 + tid] = m1 * (1.f / HID);
    S.stats[48 + tid] = m2 * (1.f / HID);
  }
  __syncthreads();
  // dx1 = rstd * (dxhat - mean(dxhat) - xhat * mean(dxhat*xhat))
  for (int i = tid; i < SPW * HID; i += NTHR) {
    int s = i / HID;
    float mu = S.stats[s], rstd = S.stats[16 + s];
    float xh = (S.x1[i] - mu) * rstd;
    S.ebuf[i] = rstd * (S.ebuf[i] - S.stats[32 + s] - xh * S.stats[48 + s]);
  }
  __syncthreads();
  // gr2 = dx1 @ h1W^T  (16 x 64, K=256) ; B from original h1W (R2 x HID)
  for (int tile = wave; tile < R2_DIM / 16; tile += NW) {
    int c0 = tile * 16;
    v8f acc = wmma_tile(S.ebuf, HID, W.h1W, HID, HID, c0, lane);
    int rn = lane & 15, sb = (lane >> 4) * 8, c = c0 + rn;
#pragma unroll
    for (int v = 0; v < 8; ++v) S.gr2[(sb + v) * R2_DIM + c] = acc[v];
  }
  __syncthreads();
}

// ---------------- decoder: ebuf = rr @ decW + decb - x_t (B from decWT) ---------
__device__ void decode_to_e(const float* __restrict__ rr, const Sm& S, const Wt& W,
                            int lane, int wave) {
  for (int tile = wave; tile < IN_DIM / 16; tile += NW) {
    int c0 = tile * 16;
    v8f acc = wmma_tile(rr, R_DIM, W.decWT, R_DIM, R_DIM, c0, lane);
    int rn = lane & 15, sb = (lane >> 4) * 8, c = c0 + rn;
#pragma unroll
    for (int v = 0; v < 8; ++v) {
      int s = sb + v;
      S.ebuf[s * IN_DIM + c] = acc[v] + W.decb[c] - S.xt[s * IN_DIM + c];
    }
  }
  __syncthreads();
}

// ---------------- one-shot weight transposes into ws (idempotent) ----------------
__global__ void transpose_weights_kernel(const float* __restrict__ decW,
                                         const float* __restrict__ h1W,
                                         const float* __restrict__ h2W,
                                         const float* __restrict__ h3W,
                                         float* __restrict__ ws) {
  float* decWT = ws + WS_WT_OFF;
  float* h1WT  = decWT + DECWT_SZ;
  float* h2WT  = h1WT + H1WT_SZ;
  float* h3WT  = h2WT + H2WT_SZ;
  const int idx = blockIdx.x * blockDim.x + threadIdx.x;
  const int stride = gridDim.x * blockDim.x;
  for (int i = idx; i < R_DIM * IN_DIM; i += stride) {       // decW (128x512) -> (512x128)
    int k = i / IN_DIM, c = i % IN_DIM;
    decWT[c * R_DIM + k] = decW[i];
  }
  for (int i = idx; i < R2_DIM * HID; i += stride) {         // h1W (64x256) -> (256x64)
    int k = i / HID, c = i % HID;
    h1WT[c * R2_DIM + k] = h1W[i];
  }
  for (int i = idx; i < HID * HID; i += stride) {            // h2W (256x256) -> (256x256)
    int k = i / HID, c = i % HID;
    h2WT[c * HID + k] = h2W[i];
  }
  for (int i = idx; i < HID * MIXK; i += stride) {           // h3W (256x32) -> (32x256)
    int k = i / MIXK, c = i % MIXK;
    h3WT[c * HID + k] = h3W[i];
  }
}

// ---------------- fused persistent kernel: one WG == 16 batch samples ----------
__global__ __launch_bounds__(NTHR) void tidhy_kernel(
    const float* __restrict__ X, const float* __restrict__ decW,
    const float* __restrict__ decb, const float* __restrict__ temporal,
    const float* __restrict__ h1W, const float* __restrict__ h1b,
    const float* __restrict__ lns, const float* __restrict__ lnb,
    const float* __restrict__ h2W, const float* __restrict__ h2b,
    const float* __restrict__ h3W, const float* __restrict__ h3b,
    float* __restrict__ out, float* __restrict__ ws) {
  extern __shared__ float smem[];
  Sm S;
  {
    float* p = smem;
    S.xt = p;    p += SPW * IN_DIM;   // 8192
    S.ebuf = p;  p += SPW * IN_DIM;   // 8192 (decoder residual / hypernet bwd scratch)
    S.r = p;     p += SPW * R_DIM;
    S.rprev = p; p += SPW * R_DIM;
    S.rhat = p;  p += SPW * R_DIM;
    S.gsp = p;   p += SPW * R_DIM;
    S.r2 = p;    p += SPW * R2_DIM;
    S.gr2 = p;   p += SPW * R2_DIM;
    S.x1 = p;    p += SPW * HID;
    S.a1 = p;    p += SPW * HID;
    S.x2d = p;   p += SPW * HID;
    S.wmix = p;  p += SPW * MIXK;
    S.x3 = p;    p += SPW * MIXK;
    S.dwm = p;   p += SPW * MIXK;
    S.stats = p; p += 64;
  }
  Wt W{decW, decb, temporal, h1W, h1b, lns, lnb, h2W, h2b, h3W, h3b,
       ws + WS_WT_OFF,
       ws + WS_WT_OFF + DECWT_SZ,
       ws + WS_WT_OFF + DECWT_SZ + H1WT_SZ,
       ws + WS_WT_OFF + DECWT_SZ + H1WT_SZ + H2WT_SZ};

  const int tid = threadIdx.x;
  const int lane = tid & 31;
  const int wave = tid >> 5;
  const int wg = blockIdx.x;
  float* P = ws + WS_P_OFF + (size_t)wg * SPW * MIXK * R_DIM;  // P[s][m][i], L2-hot

  float accL0 = 0.f, accL1 = 0.f, accL2 = 0.f;  // loss sums (tid 0 only meaningful)

  for (int i = tid; i < SPW * R_DIM; i += NTHR) { S.r[i] = 0.f; S.rprev[i] = 0.f; S.rhat[i] = 0.f; }
  for (int i = tid; i < SPW * R2_DIM; i += NTHR) S.r2[i] = 0.f;
  __syncthreads();

  for (int t = 0; t < TT; ++t) {
    // ---- load x_t = X[:, t, :]
    for (int i = tid; i < SPW * IN_DIM; i += NTHR) {
      int s = i / IN_DIM, c = i % IN_DIM;
      S.xt[i] = X[((size_t)(wg * SPW + s) * TT + t) * IN_DIM + c];
    }
    __syncthreads();

    // ---- precompute P[s][m][i] = sum_j Vk[m][i][j] * rprev[s][j]  (16x4096 GEMM, K=128)
    //      B[k=j][i] = temporal[m*R*R + i*R + j]  (k-contiguous already)
    for (int tile = wave; tile < (MIXK * R_DIM) / 16; tile += NW) {
      int c0 = tile * 16;
      int m = c0 >> 7;                       // 16-col tiles never straddle m
      int i0 = c0 & (R_DIM - 1);
      v8f acc = wmma_tile(S.rprev, R_DIM, temporal + (size_t)m * R_DIM * R_DIM,
                          R_DIM, R_DIM, i0, lane);
      int rn = lane & 15, sb = (lane >> 4) * 8, ii = i0 + rn;
#pragma unroll
      for (int v = 0; v < 8; ++v)
        P[(size_t)(sb + v) * (MIXK * R_DIM) + m * R_DIM + ii] = acc[v];
    }
    __threadfence_block();
    __syncthreads();

    // ---- warm start: r = temp_pred(rprev, r2) = hypernet(r2) . P
    hypernet_fwd(S, W, tid, lane, wave);
    for (int i = tid; i < SPW * R_DIM; i += NTHR) {
      int s = i / R_DIM, ii = i % R_DIM;
      const float* Ps = P + (size_t)s * MIXK * R_DIM + ii;
      const float* wv = S.wmix + s * MIXK;
      float acc = 0.f;
      for (int m = 0; m < MIXK; ++m) acc += wv[m] * Ps[m * R_DIM];
      S.r[i] = acc;
      S.rhat[i] = acc;
    }
    __syncthreads();

    // ---- ISTA iterations
    for (int it = 0; it < MAX_ITERS; ++it) {
      hypernet_fwd(S, W, tid, lane, wave);
      // rhat = wmix . P
      for (int i = tid; i < SPW * R_DIM; i += NTHR) {
        int s = i / R_DIM, ii = i % R_DIM;
        const float* Ps = P + (size_t)s * MIXK * R_DIM + ii;
        const float* wv = S.wmix + s * MIXK;
        float acc = 0.f;
        for (int m = 0; m < MIXK; ++m) acc += wv[m] * Ps[m * R_DIM];
        S.rhat[i] = acc;
      }
      __syncthreads();
      // e = dec(r) - x_t ; gsp = e @ decW^T (K=512, B from original decW, k-contiguous)
      decode_to_e(S.r, S, W, lane, wave);
      for (int tile = wave; tile < R_DIM / 16; tile += NW) {
        int c0 = tile * 16;
        v8f acc = wmma_tile(S.ebuf, IN_DIM, W.decW, IN_DIM, IN_DIM, c0, lane);
        int rn = lane & 15, sb = (lane >> 4) * 8, c = c0 + rn;
#pragma unroll
        for (int v = 0; v < 8; ++v) S.gsp[(sb + v) * R_DIM + c] = acc[v];
      }
      __syncthreads();
      // dE/dw[s][m] = sum_i -2 (r - rhat)[s][i] * P[s][m][i]
      for (int i = tid; i < SPW * MIXK; i += NTHR) {
        int s = i / MIXK, m = i % MIXK;
        const float* Ps = P + (size_t)s * MIXK * R_DIM + m * R_DIM;
        const float* rs = S.r + s * R_DIM;
        const float* rh = S.rhat + s * R_DIM;
        float acc = 0.f;
        for (int ii = 0; ii < R_DIM; ++ii) acc += (rs[ii] - rh[ii]) * Ps[ii];
        S.dwm[i] = -2.f * acc;
      }
      __syncthreads();
      hypernet_bwd(S, W, tid, lane, wave);
      // proximal updates (LR=0.1, lambda*LR = 0.001)
      for (int i = tid; i < SPW * R_DIM; i += NTHR) {
        float g = 2.f * S.gsp[i] + 2.f * (S.r[i] - S.rhat[i]);
        S.r[i] = soft_thresh(S.r[i] - 0.1f * g, 0.001f);
      }
      for (int i = tid; i < SPW * R2_DIM; i += NTHR)
        S.r2[i] = soft_thresh(S.r2[i] - 0.1f * S.gr2[i], 0.001f);
      __syncthreads();
    }

    // ---- losses for this timestep
    hypernet_fwd(S, W, tid, lane, wave);
    for (int i = tid; i < SPW * R_DIM; i += NTHR) {
      int s = i / R_DIM, ii = i % R_DIM;
      const float* Ps = P + (size_t)s * MIXK * R_DIM + ii;
      const float* wv = S.wmix + s * MIXK;
      float acc = 0.f;
      for (int m = 0; m < MIXK; ++m) acc += wv[m] * Ps[m * R_DIM];
      S.rhat[i] = acc;
    }
    __syncthreads();
    float p_rhat = 0.f, p_rbar = 0.f, p_temp = 0.f;
    decode_to_e(S.rhat, S, W, lane, wave);
    for (int i = tid; i < SPW * IN_DIM; i += NTHR) p_rhat += S.ebuf[i] * S.ebuf[i];
    for (int i = tid; i < SPW * R_DIM; i += NTHR) {
      float d = S.r[i] - S.rhat[i];
      p_temp += d * d;
    }
    __syncthreads();
    decode_to_e(S.r, S, W, lane, wave);
    for (int i = tid; i < SPW * IN_DIM; i += NTHR) p_rbar += S.ebuf[i] * S.ebuf[i];
    float w0 = wave_sum(p_rhat), w1 = wave_sum(p_rbar), w2 = wave_sum(p_temp);
    __syncthreads();
    if (lane == 0) {
      S.dwm[wave * 3 + 0] = w0;
      S.dwm[wave * 3 + 1] = w1;
      S.dwm[wave * 3 + 2] = w2;
    }
    __syncthreads();
    if (tid == 0) {
      for (int wv = 0; wv < NW; ++wv) {
        accL0 += S.dwm[wv * 3 + 0];
        accL1 += S.dwm[wv * 3 + 1];
        accL2 += S.dwm[wv * 3 + 2];
      }
    }
    // ---- carry: r_prev = r
    for (int i = tid; i < SPW * R_DIM; i += NTHR) S.rprev[i] = S.r[i];
    __syncthreads();
  }

  if (tid == 0) {
    ws[wg * 3 + 0] = accL0;
    ws[wg * 3 + 1] = accL1;
    ws[wg * 3 + 2] = accL2;
  }
  // ---- outputs: [losses(3)] [r_fin 256x128] [r2_fin 256x64]
  for (int i = tid; i < SPW * R_DIM; i += NTHR) {
    int s = i / R_DIM, ii = i % R_DIM;
    out[3 + (size_t)(wg * SPW + s) * R_DIM + ii] = S.r[i];
  }
  for (int i = tid; i < SPW * R2_DIM; i += NTHR) {
    int s = i / R2_DIM, j = i % R2_DIM;
    out[3 + (size_t)BB * R_DIM + (size_t)(wg * SPW + s) * R2_DIM + j] = S.r2[i];
  }
}

__global__ void finalize_kernel(float* __restrict__ out, const float* __restrict__ ws) {
  int k = threadIdx.x;
  if (k < 3) {
    float s = 0.f;
    for (int w = 0; w < NWG; ++w) s += ws[w * 3 + k];
    out[k] = s / (float)(BB * TT);   // mean over batch and time
  }
}

extern "C" void kernel_launch(void* const* d_in, const int* in_sizes, int n_in,
                              void* d_out, int out_size, void* d_ws, size_t ws_size,
                              hipStream_t stream) {
  (void)in_sizes; (void)n_in; (void)out_size; (void)ws_size;
  const float* X        = (const float*)d_in[0];
  const float* decW     = (const float*)d_in[1];
  const float* decb     = (const float*)d_in[2];
  const float* temporal = (const float*)d_in[3];
  const float* h1W      = (const float*)d_in[4];
  const float* h1b      = (const float*)d_in[5];
  const float* lns      = (const float*)d_in[6];
  const float* lnb      = (const float*)d_in[7];
  const float* h2W      = (const float*)d_in[8];
  const float* h2b      = (const float*)d_in[9];
  const float* h3W      = (const float*)d_in[10];
  const float* h3b      = (const float*)d_in[11];
  float* out = (float*)d_out;
  float* ws  = (float*)d_ws;   // needs ~4.9 MB: loss slots + P + transposed weights

  constexpr size_t SMEM_FLOATS =
      2 * SPW * IN_DIM + 4 * SPW * R_DIM + 2 * SPW * R2_DIM +
      3 * SPW * HID + 3 * SPW * MIXK + 64;                    // 40512 floats
  constexpr size_t SMEM_BYTES = SMEM_FLOATS * sizeof(float);  // ~158 KB (<320 KB/WGP)
  (void)hipFuncSetAttribute(reinterpret_cast<const void*>(tidhy_kernel),
                            hipFuncAttributeMaxDynamicSharedMemorySize,
                            (int)SMEM_BYTES);

  transpose_weights_kernel<<<64, 256, 0, stream>>>(decW, h1W, h2W, h3W, ws);
  tidhy_kernel<<<NWG, NTHR, SMEM_BYTES, stream>>>(
      X, decW, decb, temporal, h1W, h1b, lns, lnb, h2W, h2b, h3W, h3b, out, ws);
  finalize_kernel<<<1, 32, 0, stream>>>(out, ws);
}